// GNN_simple_32822140076342
// MI455X (gfx1250) — compile-verified
//
#include <hip/hip_runtime.h>

// ---------------------------------------------------------------------------
// Sizes (compile-time constants matching the reference)
// ---------------------------------------------------------------------------
#define NPFAS 20000
#define NGW   100000
#define NSW   30000
#define DFEAT 128

typedef __attribute__((ext_vector_type(2))) float v2f;
typedef __attribute__((ext_vector_type(8))) float v8f;

// ---------------------------------------------------------------------------
// Zero-fill workspace (sums + counts)
// ---------------------------------------------------------------------------
__global__ void zero_kernel(float* __restrict__ p, long n) {
  long i = (long)blockIdx.x * blockDim.x + threadIdx.x;
  long stride = (long)gridDim.x * blockDim.x;
  for (; i < n; i += stride) p[i] = 0.0f;
}

// Wr_comb = Wr_gp + Wr_sp  (128x128)
__global__ void wadd_kernel(const float* __restrict__ a, const float* __restrict__ b,
                            float* __restrict__ c, int n) {
  int i = blockIdx.x * blockDim.x + threadIdx.x;
  if (i < n) c[i] = a[i] + b[i];
}

// ---------------------------------------------------------------------------
// Edge scatter: one wave32 per edge. Each lane moves float4 (32*4 = 128 floats)
// and atomically adds into sum[dst]. Lane 0 counts the edge.
// ---------------------------------------------------------------------------
__global__ __launch_bounds__(256)
void scatter_kernel(const float* __restrict__ x, const int* __restrict__ src,
                    const int* __restrict__ dst, float* __restrict__ sum,
                    float* __restrict__ cnt, int nE) {
  int gtid = blockIdx.x * blockDim.x + threadIdx.x;
  int e = gtid >> 5;
  int lane = gtid & 31;
  if (e >= nE) return;
  int s = src[e];
  int d = dst[e];
  const float4* xs = (const float4*)(x + (size_t)s * DFEAT);
  float4 v = xs[lane];
  float* sb = sum + (size_t)d * DFEAT + lane * 4;
  unsafeAtomicAdd(sb + 0, v.x);
  unsafeAtomicAdd(sb + 1, v.y);
  unsafeAtomicAdd(sb + 2, v.z);
  unsafeAtomicAdd(sb + 3, v.w);
  if (lane == 0) unsafeAtomicAdd(cnt + d, 1.0f);
}

// ---------------------------------------------------------------------------
// WMMA helper: accumulate C += (Arow*ascale) @ B for K=128, one 16x16 tile.
//   Arow : pointer to this lane's A row (row = lane&15 of the node block)
//   B    : 128x128 row-major weight matrix, tile columns [n0, n0+16)
// A frag (16x4 f32): lane&15 = M; vgpr0 = K=2*half, vgpr1 = K=2*half+1
// B frag (4x16 f32): lane&15 = N; vgpr0 = K=2*half, vgpr1 = K=2*half+1
// ---------------------------------------------------------------------------
__device__ __forceinline__
v8f wmma_k128(const float* __restrict__ Arow, float ascale,
              const float* __restrict__ B, int n0, int lane, v8f c) {
  const int hf = lane >> 4;
  const int l  = lane & 15;
  const int kk = 2 * hf;
#pragma unroll 4
  for (int k0 = 0; k0 < DFEAT; k0 += 4) {
    int k = k0 + kk;
    v2f a, b;
    a.x = Arow[k]     * ascale;
    a.y = Arow[k + 1] * ascale;
    b.x = B[(size_t)k       * DFEAT + n0 + l];
    b.y = B[(size_t)(k + 1) * DFEAT + n0 + l];
    c = __builtin_amdgcn_wmma_f32_16x16x4_f32(false, a, false, b,
                                              (short)0, c, false, false);
  }
  return c;
}

// ---------------------------------------------------------------------------
// gw / sw node update: h = relu(mean@Wl + bl + x@Wr); out = h @ Wout + bout.
// One wave per 16-node block; loops 8 column tiles; head folded into dot
// partials reduced with __shfl_xor (h never hits memory).
// ---------------------------------------------------------------------------
__global__ __launch_bounds__(128)
void node_score_kernel(const float* __restrict__ sum, const float* __restrict__ cnt,
                       const float* __restrict__ xd,
                       const float* __restrict__ Wl, const float* __restrict__ bl,
                       const float* __restrict__ Wr,
                       const float* __restrict__ Wout, const float* __restrict__ bout,
                       float* __restrict__ out, int nNodes) {
  const int waveIb = threadIdx.x >> 5;
  const int lane   = threadIdx.x & 31;
  const int rowBase = (blockIdx.x * 4 + waveIb) * 16;
  if (rowBase >= nNodes) return;
  const int hf = lane >> 4;
  const int l  = lane & 15;
  const int row = rowBase + l;

  const float invc = 1.0f / fmaxf(cnt[row], 1.0f);
  const float* ArowS = sum + (size_t)row * DFEAT;
  const float* ArowX = xd  + (size_t)row * DFEAT;

  float p[8];
#pragma unroll
  for (int i = 0; i < 8; ++i) p[i] = 0.0f;

  for (int t = 0; t < 8; ++t) {
    const int n0 = t * 16;
    v8f c = {};
    c = wmma_k128(ArowS, invc, Wl, n0, lane, c);   // mean @ Wl
    c = wmma_k128(ArowX, 1.0f, Wr, n0, lane, c);   // x_dst @ Wr
    const float blv = bl[n0 + l];
    const float wv  = Wout[n0 + l];
#pragma unroll
    for (int i = 0; i < 8; ++i) {
      float h = fmaxf(c[i] + blv, 0.0f);           // ReLU
      p[i] += h * wv;                              // head partial
    }
  }
  // reduce dot partials within each 16-lane half
#pragma unroll
  for (int i = 0; i < 8; ++i) {
    float v = p[i];
    v += __shfl_xor(v, 1);
    v += __shfl_xor(v, 2);
    v += __shfl_xor(v, 4);
    v += __shfl_xor(v, 8);
    p[i] = v;
  }
  if (l == 0) {
    const float b0 = bout[0];
#pragma unroll
    for (int i = 0; i < 8; ++i)
      out[rowBase + 8 * hf + i] = p[i] + b0;
  }
}

// ---------------------------------------------------------------------------
// pfas node update (two incoming relations, HeteroConv sum):
// h = relu(mean_gp@Wl_gp + bl_gp + mean_sp@Wl_sp + bl_sp + x@(Wr_gp+Wr_sp))
// ---------------------------------------------------------------------------
__global__ __launch_bounds__(128)
void pfas_kernel(const float* __restrict__ sumGp, const float* __restrict__ cntGp,
                 const float* __restrict__ sumSp, const float* __restrict__ cntSp,
                 const float* __restrict__ x,
                 const float* __restrict__ WlGp, const float* __restrict__ blGp,
                 const float* __restrict__ WlSp, const float* __restrict__ blSp,
                 const float* __restrict__ WrComb,
                 float* __restrict__ hOut, int nNodes) {
  const int waveIb = threadIdx.x >> 5;
  const int lane   = threadIdx.x & 31;
  const int rowBase = (blockIdx.x * 4 + waveIb) * 16;
  if (rowBase >= nNodes) return;
  const int hf = lane >> 4;
  const int l  = lane & 15;
  const int row = rowBase + l;

  const float invG = 1.0f / fmaxf(cntGp[row], 1.0f);
  const float invS = 1.0f / fmaxf(cntSp[row], 1.0f);
  const float* ArowG = sumGp + (size_t)row * DFEAT;
  const float* ArowS = sumSp + (size_t)row * DFEAT;
  const float* ArowX = x     + (size_t)row * DFEAT;

  for (int t = 0; t < 8; ++t) {
    const int n0 = t * 16;
    v8f c = {};
    c = wmma_k128(ArowG, invG, WlGp,   n0, lane, c);
    c = wmma_k128(ArowS, invS, WlSp,   n0, lane, c);
    c = wmma_k128(ArowX, 1.0f, WrComb, n0, lane, c);
    const float blv = blGp[n0 + l] + blSp[n0 + l];
#pragma unroll
    for (int i = 0; i < 8; ++i) {
      float h = fmaxf(c[i] + blv, 0.0f);
      hOut[(size_t)(rowBase + 8 * hf + i) * DFEAT + n0 + l] = h;
    }
  }
}

// ---------------------------------------------------------------------------
extern "C" void kernel_launch(void* const* d_in, const int* in_sizes, int n_in,
                              void* d_out, int out_size, void* d_ws, size_t ws_size,
                              hipStream_t stream) {
  const float* x_pf = (const float*)d_in[0];
  const float* x_gw = (const float*)d_in[1];
  const float* x_sw = (const float*)d_in[2];
  const int* pg_src = (const int*)d_in[3];
  const int* pg_dst = (const int*)d_in[4];
  const int* gp_src = (const int*)d_in[5];
  const int* gp_dst = (const int*)d_in[6];
  const int* ps_src = (const int*)d_in[7];
  const int* ps_dst = (const int*)d_in[8];
  const int* sp_src = (const int*)d_in[9];
  const int* sp_dst = (const int*)d_in[10];
  const float* Wl_pg = (const float*)d_in[11];
  const float* bl_pg = (const float*)d_in[12];
  const float* Wr_pg = (const float*)d_in[13];
  const float* Wl_gp = (const float*)d_in[14];
  const float* bl_gp = (const float*)d_in[15];
  const float* Wr_gp = (const float*)d_in[16];
  const float* Wl_ps = (const float*)d_in[17];
  const float* bl_ps = (const float*)d_in[18];
  const float* Wr_ps = (const float*)d_in[19];
  const float* Wl_sp = (const float*)d_in[20];
  const float* bl_sp = (const float*)d_in[21];
  const float* Wr_sp = (const float*)d_in[22];
  const float* W_gw  = (const float*)d_in[23];
  const float* b_gw  = (const float*)d_in[24];
  const float* W_sw  = (const float*)d_in[25];
  const float* b_sw  = (const float*)d_in[26];

  const int E_PG = in_sizes[3];
  const int E_GP = in_sizes[5];
  const int E_PS = in_sizes[7];
  const int E_SP = in_sizes[9];

  // ---- workspace layout (contiguous f32) ----
  float* ws = (float*)d_ws;
  float* sum_pg = ws;  ws += (size_t)NGW   * DFEAT;
  float* sum_ps = ws;  ws += (size_t)NSW   * DFEAT;
  float* sum_gp = ws;  ws += (size_t)NPFAS * DFEAT;
  float* sum_sp = ws;  ws += (size_t)NPFAS * DFEAT;
  float* cnt_pg = ws;  ws += NGW;
  float* cnt_ps = ws;  ws += NSW;
  float* cnt_gp = ws;  ws += NPFAS;
  float* cnt_sp = ws;  ws += NPFAS;
  float* WrComb = ws;  ws += DFEAT * DFEAT;

  const long zeroN = (long)DFEAT * (NGW + NSW + 2L * NPFAS) + (NGW + NSW + 2L * NPFAS);
  zero_kernel<<<2048, 256, 0, stream>>>((float*)d_ws, zeroN);
  wadd_kernel<<<(DFEAT * DFEAT + 255) / 256, 256, 0, stream>>>(Wr_gp, Wr_sp, WrComb,
                                                               DFEAT * DFEAT);

  // ---- scatter phase: one wave32 per edge ----
  {
    int thr = 256;  // 8 waves / block
    int b;
    b = (E_PG * 32 + thr - 1) / thr;
    scatter_kernel<<<b, thr, 0, stream>>>(x_pf, pg_src, pg_dst, sum_pg, cnt_pg, E_PG);
    b = (E_PS * 32 + thr - 1) / thr;
    scatter_kernel<<<b, thr, 0, stream>>>(x_pf, ps_src, ps_dst, sum_ps, cnt_ps, E_PS);
    b = (E_GP * 32 + thr - 1) / thr;
    scatter_kernel<<<b, thr, 0, stream>>>(x_gw, gp_src, gp_dst, sum_gp, cnt_gp, E_GP);
    b = (E_SP * 32 + thr - 1) / thr;
    scatter_kernel<<<b, thr, 0, stream>>>(x_sw, sp_src, sp_dst, sum_sp, cnt_sp, E_SP);
  }

  // ---- node update phase (WMMA) ----
  float* h_pf   = (float*)d_out;
  float* out_gw = h_pf + (size_t)NPFAS * DFEAT;
  float* out_sw = out_gw + NGW;

  // 4 waves per 128-thread block, one 16-node group per wave
  node_score_kernel<<<(NGW / 16 + 3) / 4, 128, 0, stream>>>(
      sum_pg, cnt_pg, x_gw, Wl_pg, bl_pg, Wr_pg, W_gw, b_gw, out_gw, NGW);
  node_score_kernel<<<(NSW / 16 + 3) / 4, 128, 0, stream>>>(
      sum_ps, cnt_ps, x_sw, Wl_ps, bl_ps, Wr_ps, W_sw, b_sw, out_sw, NSW);
  pfas_kernel<<<(NPFAS / 16 + 3) / 4, 128, 0, stream>>>(
      sum_gp, cnt_gp, sum_sp, cnt_sp, x_pf, Wl_gp, bl_gp, Wl_sp, bl_sp, WrComb,
      h_pf, NPFAS);
}